// BlockwiseWTA_46995532153184
// MI455X (gfx1250) — compile-verified
//
#include <hip/hip_runtime.h>
#include <hip/hip_bf16.h>

#define EMBED      4096
#define NBLK       8
#define BS         512     // floats per block
#define TOPK_K     32
#define WAVES      8
#define THREADS    256
#define ROWS_PER_WG 8

typedef float v4f __attribute__((ext_vector_type(4)));  // native vector for NT stores

// ---- wave32 helpers -------------------------------------------------------
__device__ __forceinline__ int wave_sum(int v) {
#pragma unroll
  for (int m = 16; m >= 1; m >>= 1) v += __shfl_xor(v, m, 32);
  return v;  // full 32-lane sum, uniform across wave
}

__device__ __forceinline__ int wave_excl_scan(int v, int lane) {
  int s = v;
#pragma unroll
  for (int d = 1; d < 32; d <<= 1) {
    int t = __shfl_up(s, d, 32);
    if (lane >= d) s += t;
  }
  return s - v;  // exclusive prefix sum over lanes
}

// ---------------------------------------------------------------------------
// One wave handles one 512-float block: async-DMA it to LDS (ASYNCcnt path),
// compute the exact 32nd-largest value via bitwise binary search on
// order-preserving keys, then stream value-or-zero back out with NT stores.
// Double-buffered over 8 rows so the DMA of row r+1 overlaps row r's math.
// ---------------------------------------------------------------------------
__global__ __launch_bounds__(THREADS) void blockwise_wta_kernel(
    const float* __restrict__ x, float* __restrict__ out, int batch) {
  __shared__ float smem[2 * WAVES * BS];  // 32 KB, double buffered

  const int wave = threadIdx.x >> 5;
  const int lane = threadIdx.x & 31;
  const int row0 = blockIdx.x * ROWS_PER_WG;
  if (row0 >= batch) return;

  // Low 32 bits of a flat pointer into the LDS aperture == LDS byte offset.
  const uint32_t lds_base = (uint32_t)(uintptr_t)(&smem[0]);

  // Issue 4 async B128 loads: 32 lanes x 16B x 4 = 2 KB (one block).
  // IOFFSET is added to BOTH the global and the LDS address (ISA 15.18),
  // so a single per-lane base pair covers all four chunks.
  auto issue_async = [&](int row, int buf) {
    uint32_t voff = (uint32_t)row * (EMBED * 4u) +
                    (uint32_t)wave * (BS * 4u) + (uint32_t)lane * 16u;
    uint32_t ldsoff = lds_base + (uint32_t)buf * (WAVES * BS * 4u) +
                      (uint32_t)wave * (BS * 4u) + (uint32_t)lane * 16u;
    asm volatile(
        "global_load_async_to_lds_b128 %0, %1, %2 offset:0\n\t"
        "global_load_async_to_lds_b128 %0, %1, %2 offset:512\n\t"
        "global_load_async_to_lds_b128 %0, %1, %2 offset:1024\n\t"
        "global_load_async_to_lds_b128 %0, %1, %2 offset:1536"
        :
        : "v"(ldsoff), "v"(voff), "s"(x)
        : "memory");
  };

  issue_async(row0, 0);

  for (int r = 0; r < ROWS_PER_WG; ++r) {
    const int row = row0 + r;
    if (row >= batch) break;

    const bool more = (r + 1 < ROWS_PER_WG) && (row + 1 < batch);
    if (more) {
      issue_async(row + 1, (r + 1) & 1);
      // current buffer's 4 loads done; next buffer's 4 still in flight
      asm volatile("s_wait_asynccnt 0x4" ::: "memory");
    } else {
      asm volatile("s_wait_asynccnt 0x0" ::: "memory");
    }

    const int buf = r & 1;
    const v4f* sp = (const v4f*)(&smem[buf * (WAVES * BS) + wave * BS]);

    float    val[16];
    uint32_t key[16];
#pragma unroll
    for (int c = 0; c < 4; ++c) {            // ds_load_b128 x4
      v4f v4 = sp[c * 32 + lane];
      val[c * 4 + 0] = v4.x; val[c * 4 + 1] = v4.y;
      val[c * 4 + 2] = v4.z; val[c * 4 + 3] = v4.w;
    }
#pragma unroll
    for (int i = 0; i < 16; ++i) {
      uint32_t u = __float_as_uint(val[i]);
      key[i] = (u & 0x80000000u) ? ~u : (u | 0x80000000u);  // order-preserving
    }

    // Exact K-th largest key via greedy bit search (monotone in the key).
    uint32_t thr = 0;
#pragma unroll
    for (int bit = 31; bit >= 0; --bit) {
      const uint32_t trial = thr | (1u << bit);
      int c = 0;
#pragma unroll
      for (int i = 0; i < 16; ++i) c += (key[i] >= trial) ? 1 : 0;
      c = wave_sum(c);
      if (c >= TOPK_K) thr = trial;  // uniform branch
    }

    // Tie quota: keep exactly K elements total.
    int cg = 0, ce = 0;
#pragma unroll
    for (int i = 0; i < 16; ++i) {
      cg += (key[i] > thr) ? 1 : 0;
      ce += (key[i] == thr) ? 1 : 0;
    }
    const int G     = wave_sum(cg);
    const int quota = TOPK_K - G;                // >= 1 by construction
    const int excl  = wave_excl_scan(ce, lane);
    int keep = quota - excl;
    keep = keep < 0 ? 0 : (keep > ce ? ce : keep);

    int eseen = 0;
    float o[16];
#pragma unroll
    for (int i = 0; i < 16; ++i) {
      if (key[i] > thr)                       { o[i] = val[i]; }
      else if (key[i] == thr && eseen < keep) { o[i] = val[i]; ++eseen; }
      else                                    { o[i] = 0.0f; }
    }

    // Output is write-once streamed (512 MB total traffic > 192 MB L2):
    // non-temporal B128 stores keep L2 capacity for the async input stream.
    v4f* op = (v4f*)(out + (size_t)row * EMBED + wave * BS);
#pragma unroll
    for (int c = 0; c < 4; ++c) {            // global_store_b128 th:NT x4
      v4f v;
      v.x = o[c * 4 + 0]; v.y = o[c * 4 + 1];
      v.z = o[c * 4 + 2]; v.w = o[c * 4 + 3];
      __builtin_nontemporal_store(v, op + c * 32 + lane);
    }
  }
}

extern "C" void kernel_launch(void* const* d_in, const int* in_sizes, int n_in,
                              void* d_out, int out_size, void* d_ws, size_t ws_size,
                              hipStream_t stream) {
  const float* x = (const float*)d_in[0];
  float* out = (float*)d_out;
  const int batch = in_sizes[0] / EMBED;                    // 16384
  const int nwg = (batch + ROWS_PER_WG - 1) / ROWS_PER_WG;  // 2048
  blockwise_wta_kernel<<<dim3(nwg), dim3(THREADS), 0, stream>>>(x, out, batch);
}